// Net_64415919505441
// MI455X (gfx1250) — compile-verified
//
#include <hip/hip_runtime.h>
#include <hip/hip_bf16.h>

typedef __attribute__((ext_vector_type(2))) float v2f;
typedef __attribute__((ext_vector_type(8))) float v8f;

#define WPAD 66          // row pad (dwords) -> bank-conflict-free transposed reads
#define EPS 1e-5f

__device__ __forceinline__ float leaky(float x) { return x > 0.0f ? x : 0.01f * x; }

// ---------------------------------------------------------------- utility fills
__global__ void fill_f32_kernel(float* __restrict__ p, float v, int n) {
  int i = blockIdx.x * blockDim.x + threadIdx.x;
  if (i < n) p[i] = v;
}
__global__ void fill_u32_kernel(unsigned* __restrict__ p, unsigned v, int n) {
  int i = blockIdx.x * blockDim.x + threadIdx.x;
  if (i < n) p[i] = v;
}

// ---------------------------------------------------------------- degree / norm
__global__ void deg_kernel(const int* __restrict__ row, float* __restrict__ deg, int E) {
  int e = blockIdx.x * blockDim.x + threadIdx.x;
  if (e < E) atomicAdd(&deg[row[e]], 1.0f);
}
__global__ void rsqrt_kernel(float* __restrict__ d, int n) {
  int i = blockIdx.x * blockDim.x + threadIdx.x;
  if (i < n) d[i] = rsqrtf(d[i]);
}

// ---------------------------------------------------------------- dense MLP (WMMA)
// H = leaky(BN(X @ W1 + b1)) @ W2 + b2     (optionally leaky applied to X on load)
// 128 threads = 4 waves; each wave owns a 16-node tile; K=64 via 16 x (16x16x4) WMMA.
__global__ __launch_bounds__(128)
void dense_mlp_kernel(const float* __restrict__ X,
                      const float* __restrict__ W1, const float* __restrict__ B1,
                      const float* __restrict__ G,  const float* __restrict__ BE,
                      const float* __restrict__ MU, const float* __restrict__ VAR,
                      const float* __restrict__ W2, const float* __restrict__ B2,
                      float* __restrict__ H, int n, int leaky_in)
{
  __shared__ float w1t[64 * WPAD];        // w1t[c][k] = W1[k][c]
  __shared__ float w2t[64 * WPAD];
  __shared__ float hS[4][16 * WPAD];      // per-wave 16x64 intermediate (row-major)

  const int tid = threadIdx.x;
  for (int idx = tid; idx < 64 * 64; idx += 128) {
    const int k = idx >> 6, c = idx & 63;
    w1t[c * WPAD + k] = W1[idx];
    w2t[c * WPAD + k] = W2[idx];
  }
  __syncthreads();

  const int wave = tid >> 5, lane = tid & 31;
  const int lm = lane & 15, lh = lane >> 4;      // lh: K pair select (A/B frag layout)
  const int tb = (blockIdx.x * 4 + wave) * 16;   // 16 | N so tiles are all-or-nothing
  if (tb >= n) return;

  // A fragments (16 k-steps) straight from global X
  v2f a[16];
  const float* xr = X + (size_t)(tb + lm) * 64 + lh * 2;
  #pragma unroll
  for (int kk = 0; kk < 16; ++kk) {
    float x0 = xr[kk * 4 + 0];
    float x1 = xr[kk * 4 + 1];
    if (leaky_in) { x0 = leaky(x0); x1 = leaky(x1); }
    a[kk] = (v2f){x0, x1};
  }

  float* myS = hS[wave];

  // ---- GEMM1 + bias + BN + leaky, result into per-wave LDS scratch ----
  #pragma unroll
  for (int nt = 0; nt < 4; ++nt) {
    v8f c = {};
    #pragma unroll
    for (int kk = 0; kk < 16; ++kk) {
      const v2f b = *(const v2f*)&w1t[(nt * 16 + lm) * WPAD + kk * 4 + lh * 2];
      c = __builtin_amdgcn_wmma_f32_16x16x4_f32(false, a[kk], false, b,
                                                (short)0, c, false, false);
    }
    const int col = nt * 16 + lm;                 // BN params depend only on col
    const float sc = rsqrtf(VAR[col] + EPS) * G[col];
    const float mu = MU[col], be = BE[col], bb = B1[col];
    #pragma unroll
    for (int r = 0; r < 8; ++r) {
      float v = c[r] + bb;
      v = (v - mu) * sc + be;
      v = leaky(v);
      myS[(r + lh * 8) * WPAD + col] = v;         // row m = r + 8*lh
    }
  }
  __builtin_amdgcn_wave_barrier();                // DS in-order within wave

  // ---- GEMM2: A fragments from LDS scratch ----
  v2f a2[16];
  #pragma unroll
  for (int kk = 0; kk < 16; ++kk)
    a2[kk] = *(const v2f*)&myS[lm * WPAD + kk * 4 + lh * 2];

  #pragma unroll
  for (int nt = 0; nt < 4; ++nt) {
    v8f c = {};
    #pragma unroll
    for (int kk = 0; kk < 16; ++kk) {
      const v2f b = *(const v2f*)&w2t[(nt * 16 + lm) * WPAD + kk * 4 + lh * 2];
      c = __builtin_amdgcn_wmma_f32_16x16x4_f32(false, a2[kk], false, b,
                                                (short)0, c, false, false);
    }
    const int col = nt * 16 + lm;
    const float bb = B2[col];
    #pragma unroll
    for (int r = 0; r < 8; ++r)
      H[(size_t)(tb + r + lh * 8) * 64 + col] = c[r] + bb;
  }
}

// ---------------------------------------------------------------- edge gather/scatter
// 16 lanes per edge, 4 dims per lane (float4 gather, 4x f32 atomicAdd scatter)
__global__ __launch_bounds__(256)
void edge_agg_kernel(const int* __restrict__ ei, const float* __restrict__ eattr,
                     const float* __restrict__ dis, const float* __restrict__ Hin,
                     float* __restrict__ Aout, int E, int N)
{
  const int t = blockIdx.x * 256 + threadIdx.x;
  const int e = t >> 4;
  const int q = t & 15;
  const int ET = E + N;
  if (e >= ET) return;

  int r, c; float ea;
  if (e < E) { r = ei[e]; c = ei[E + e]; ea = eattr[e]; }
  else       { r = e - E; c = r;          ea = 1.0f;    }   // self loop

  const float coef = dis[r] * dis[c] * ea;
  const float4 hv = *(const float4*)(Hin + (size_t)r * 64 + q * 4);
  float* outp = Aout + (size_t)c * 64 + q * 4;
  atomicAdd(outp + 0, coef * hv.x);
  atomicAdd(outp + 1, coef * hv.y);
  atomicAdd(outp + 2, coef * hv.z);
  atomicAdd(outp + 3, coef * hv.w);
}

// ---------------------------------------------------------------- segment_max pool
__global__ __launch_bounds__(256)
void pool_kernel(const float* __restrict__ A, const int* __restrict__ batch,
                 unsigned* __restrict__ pool, int n)
{
  const int t = blockIdx.x * 256 + threadIdx.x;
  const int node = t >> 6;
  const int d = t & 63;
  if (node >= n) return;
  float v = leaky(A[(size_t)node * 64 + d]);
  unsigned bits = __float_as_uint(v);
  unsigned enc = (bits & 0x80000000u) ? ~bits : (bits | 0x80000000u);  // order-preserving
  atomicMax(&pool[batch[node] * 64 + d], enc);
}

// ---------------------------------------------------------------- head (tiny)
__global__ __launch_bounds__(128)
void head_kernel(const unsigned* __restrict__ pool, const float* __restrict__ energy,
                 const float* __restrict__ e_w, const float* __restrict__ e_b,
                 const float* __restrict__ e_g, const float* __restrict__ e_be,
                 const float* __restrict__ e_mu, const float* __restrict__ e_var,
                 const float* __restrict__ l_w1, const float* __restrict__ l_b1,
                 const float* __restrict__ l_g, const float* __restrict__ l_be,
                 const float* __restrict__ l_mu, const float* __restrict__ l_var,
                 const float* __restrict__ l_w2, const float* __restrict__ l_b2,
                 float* __restrict__ out, int B)
{
  const int b = blockIdx.x * blockDim.x + threadIdx.x;
  if (b >= B) return;

  float ee[8];
  #pragma unroll
  for (int j = 0; j < 8; ++j) {
    float s = e_b[j];
    for (int k = 0; k < 21; ++k) s += energy[b * 21 + k] * e_w[k * 8 + j];
    s = (s - e_mu[j]) * rsqrtf(e_var[j] + EPS) * e_g[j] + e_be[j];
    ee[j] = leaky(s);
  }
  float gm[64];
  #pragma unroll
  for (int d = 0; d < 64; ++d) {
    const unsigned u = pool[b * 64 + d];
    gm[d] = (u & 0x80000000u) ? __uint_as_float(u ^ 0x80000000u) : __uint_as_float(~u);
  }
  float z = l_b2[0];
  for (int j = 0; j < 64; ++j) {
    float s = l_b1[j];
    for (int d = 0; d < 64; ++d) s += gm[d] * l_w1[d * 64 + j];
    #pragma unroll
    for (int k = 0; k < 8; ++k) s += ee[k] * l_w1[(64 + k) * 64 + j];
    s = (s - l_mu[j]) * rsqrtf(l_var[j] + EPS) * l_g[j] + l_be[j];
    z += leaky(s) * l_w2[j];
  }
  out[b] = -(z > 0.0f ? z : 0.0f);
}

// ================================================================ launch
extern "C" void kernel_launch(void* const* d_in, const int* in_sizes, int n_in,
                              void* d_out, int out_size, void* d_ws, size_t ws_size,
                              hipStream_t stream)
{
  (void)n_in; (void)out_size; (void)ws_size;
  // setup_inputs() dict order
  const float* x          = (const float*)d_in[0];
  const int*   edge_index = (const int*)  d_in[1];   // (2, E)
  const float* edge_attr  = (const float*)d_in[2];
  const int*   batch      = (const int*)  d_in[3];
  const float* energy     = (const float*)d_in[4];
  const float* m0_w1 = (const float*)d_in[5];  const float* m0_b1 = (const float*)d_in[6];
  const float* m0_g  = (const float*)d_in[7];  const float* m0_be = (const float*)d_in[8];
  const float* m0_mu = (const float*)d_in[9];  const float* m0_var= (const float*)d_in[10];
  const float* m0_w2 = (const float*)d_in[11]; const float* m0_b2 = (const float*)d_in[12];
  const float* m1_w1 = (const float*)d_in[13]; const float* m1_b1 = (const float*)d_in[14];
  const float* m1_g  = (const float*)d_in[15]; const float* m1_be = (const float*)d_in[16];
  const float* m1_mu = (const float*)d_in[17]; const float* m1_var= (const float*)d_in[18];
  const float* m1_w2 = (const float*)d_in[19]; const float* m1_b2 = (const float*)d_in[20];
  const float* e_w   = (const float*)d_in[21]; const float* e_b   = (const float*)d_in[22];
  const float* e_g   = (const float*)d_in[23]; const float* e_be  = (const float*)d_in[24];
  const float* e_mu  = (const float*)d_in[25]; const float* e_var = (const float*)d_in[26];
  const float* l_w1  = (const float*)d_in[27]; const float* l_b1  = (const float*)d_in[28];
  const float* l_g   = (const float*)d_in[29]; const float* l_be  = (const float*)d_in[30];
  const float* l_mu  = (const float*)d_in[31]; const float* l_var = (const float*)d_in[32];
  const float* l_w2  = (const float*)d_in[33]; const float* l_b2  = (const float*)d_in[34];

  const int N = in_sizes[0] / 64;
  const int E = in_sizes[2];
  const int B = in_sizes[4] / 21;

  // workspace partition
  float*    bufH = (float*)d_ws;                       // N x 64
  float*    bufA = bufH + (size_t)N * 64;              // N x 64
  float*    dis  = bufA + (size_t)N * 64;              // N   (deg -> deg^-1/2)
  unsigned* pool = (unsigned*)(dis + N);               // B x 64 encoded floats

  const int T = 256;
  // 1) degree (self-loop contributes 1) -> deg^-1/2
  fill_f32_kernel<<<(N + T - 1) / T, T, 0, stream>>>(dis, 1.0f, N);
  deg_kernel<<<(E + T - 1) / T, T, 0, stream>>>(edge_index, dis, E);
  rsqrt_kernel<<<(N + T - 1) / T, T, 0, stream>>>(dis, N);

  const int dgrid = (N + 63) / 64;
  const int feat  = N * 64;

  // 2) layer 0 dense MLP (WMMA)
  dense_mlp_kernel<<<dgrid, 128, 0, stream>>>(x, m0_w1, m0_b1, m0_g, m0_be, m0_mu,
                                              m0_var, m0_w2, m0_b2, bufH, N, 0);
  // 3) layer 0 aggregate
  fill_f32_kernel<<<(feat + T - 1) / T, T, 0, stream>>>(bufA, 0.0f, feat);
  edge_agg_kernel<<<(((E + N) * 16) + T - 1) / T, T, 0, stream>>>(
      edge_index, edge_attr, dis, bufH, bufA, E, N);

  // 4) layer 1 dense MLP (leaky applied to aggregated input)
  dense_mlp_kernel<<<dgrid, 128, 0, stream>>>(bufA, m1_w1, m1_b1, m1_g, m1_be, m1_mu,
                                              m1_var, m1_w2, m1_b2, bufH, N, 1);
  // 5) layer 1 aggregate
  fill_f32_kernel<<<(feat + T - 1) / T, T, 0, stream>>>(bufA, 0.0f, feat);
  edge_agg_kernel<<<(((E + N) * 16) + T - 1) / T, T, 0, stream>>>(
      edge_index, edge_attr, dis, bufH, bufA, E, N);

  // 6) global max pool (leaky fused), encoded-uint atomicMax; init = enc(-inf)
  fill_u32_kernel<<<(B * 64 + T - 1) / T, T, 0, stream>>>(pool, 0x007FFFFFu, B * 64);
  pool_kernel<<<(feat + T - 1) / T, T, 0, stream>>>(bufA, batch, pool, N);

  // 7) head -> d_out [B]
  head_kernel<<<(B + 127) / 128, 128, 0, stream>>>(
      pool, energy, e_w, e_b, e_g, e_be, e_mu, e_var,
      l_w1, l_b1, l_g, l_be, l_mu, l_var, l_w2, l_b2, (float*)d_out, B);
}